// MaskPath_7335804141970
// MI455X (gfx1250) — compile-verified
//
#include <hip/hip_runtime.h>
#include <hip/hip_bf16.h>
#include <cstdint>

// MaskPath (random-walk edge masking) for MI455X / gfx1250.
// Memory-bound pipeline: 40-bit LSD radix sort + scans + gathers.
// CDNA5-specific paths used:
//   - v_wmma_f32_16x16x4_f32 triangular-matmul block prefix scan (exact: all
//     partial sums are integers < 2^24, f32 WMMA is exact there)
//   - global_load_async_to_lds_b64 + s_wait_asynccnt   (radix histogram staging)
//   - tensor_load_to_lds (TDM) + s_wait_tensorcnt      (radix scatter tile fetch)
//   - wave32 ballot ranking (__builtin_amdgcn_ballot_w32) for stable scatter
//   - __builtin_prefetch -> global_prefetch_b8

typedef unsigned int       u32;
typedef unsigned long long u64;
typedef unsigned char      u8;

typedef __attribute__((ext_vector_type(4))) u32   su4;
typedef __attribute__((ext_vector_type(8))) u32   su8;
typedef __attribute__((ext_vector_type(2))) float v2f;
typedef __attribute__((ext_vector_type(8))) float v8f;

#define TPB   256
#define KPB   1024        // keys per block (TPB * 4)
#define RADIX 256

static __device__ __forceinline__ u32 rfl(u32 x) {
  return (u32)__builtin_amdgcn_readfirstlane((int)x);
}

static __device__ __forceinline__ u32 hmix(u32 x) {
  x ^= x >> 16; x *= 0x7feb352du;
  x ^= x >> 15; x *= 0x846ca68bu;
  x ^= x >> 16; return x;
}

// 256-wide inclusive scan via WMMA triangular matmul.
// X (16x16, row-major over thread index) ; P = X * U (U upper-tri ones) gives
// row-wise prefixes via 4 accumulating V_WMMA_F32_16X16X4_F32; cross-row base
// added per-thread. Exact for integer values with sums < 2^24 (true here:
// grand totals <= E = 16e6 < 2^24). Leaves inclusive values in s[]; total in
// s[TPB-1].
static __device__ __forceinline__ u32 block_incl_scan(u32 v, u32* s, float* sp) {
  int t = threadIdx.x;
  s[t] = v;
  __syncthreads();
  if (t < 32) {                       // wave 0 does the 16x16 matmul
    int lane = t;
    int n  = lane & 15;               // doubles as A's M and B's N
    int hi = (lane >= 16) ? 1 : 0;
    v8f c = {};
    #pragma unroll
    for (int kc = 0; kc < 16; kc += 4) {
      // A 16x4 f32: VGPR0 = K=kc+0 (lanes 0-15) / K=kc+2 (lanes 16-31);
      //             VGPR1 = K=kc+1 / K=kc+3          (ISA 7.12.2)
      int k0 = kc + (hi ? 2 : 0);
      int k1 = kc + (hi ? 3 : 1);
      v2f a, b;
      a.x = (float)s[n * 16 + k0];
      a.y = (float)s[n * 16 + k1];
      // B 4x16 ones-upper-triangular chunk: U[k][n] = (k <= n)
      b.x = (k0 <= n) ? 1.0f : 0.0f;
      b.y = (k1 <= n) ? 1.0f : 0.0f;
      c = __builtin_amdgcn_wmma_f32_16x16x4_f32(false, a, false, b,
                                                (short)0, c, false, false);
    }
    // D layout: VGPR vv -> M = vv (lanes 0-15) / vv+8 (lanes 16-31), N = lane&15
    #pragma unroll
    for (int vv = 0; vv < 8; vv++) {
      int M = vv + (hi ? 8 : 0);
      sp[M * 16 + n] = c[vv];
    }
  }
  __syncthreads();
  float base = 0.0f;
  int r = t >> 4;
  for (int m = 0; m < r; ++m) base += sp[m * 16 + 15];   // row totals prefix
  u32 incl = (u32)(base + sp[t]);
  s[t] = incl;
  __syncthreads();
  return incl;
}

// ---------------------------------------------------------------- utility
__global__ void k_zero(u32* __restrict__ p, int n) {
  int i = blockIdx.x * TPB + threadIdx.x;
  if (i < n) p[i] = 0u;
}

// key = (row << 20) | col  (row,col < 2^20 since N = 1e6)
__global__ void k_pack(const int* __restrict__ row, const int* __restrict__ col,
                       u64* __restrict__ keys, int E) {
  int i = blockIdx.x * TPB + threadIdx.x;
  if (i < E) {
    if (i + 8192 < E) {                       // global_prefetch_b8
      __builtin_prefetch(&row[i + 8192], 0, 0);
      __builtin_prefetch(&col[i + 8192], 0, 0);
    }
    keys[i] = ((u64)(u32)row[i] << 20) | (u64)(u32)col[i];
  }
}

// ------------------------------------------------------ radix: histogram
// Stage the 8KB key tile into LDS with async global->LDS DMA (ASYNCcnt path).
__global__ void k_radix_hist(const u64* __restrict__ keys, u32* __restrict__ bh,
                             u32* __restrict__ dt, int E, int shift, int numBlocks) {
  __shared__ u64 tile[KPB];
  __shared__ u32 hist[RADIX];
  int t = threadIdx.x, blk = blockIdx.x;
  int base = blk * KPB;
  hist[t] = 0u;
  #pragma unroll
  for (int j = 0; j < 4; j++) {
    int idx = j * TPB + t;
    int i = base + idx;
    if (i < E) {
      u32 lds_off = (u32)(uintptr_t)(&tile[idx]);     // low 32 bits = LDS byte addr
      u64 ga      = (u64)(uintptr_t)(keys + i);
      asm volatile("global_load_async_to_lds_b64 %0, %1, off"
                   :: "v"(lds_off), "v"(ga) : "memory");
    }
  }
  asm volatile("s_wait_asynccnt 0x0" ::: "memory");
  __syncthreads();
  #pragma unroll
  for (int j = 0; j < 4; j++) {
    int idx = j * TPB + t;
    if (base + idx < E) {
      u32 d = (u32)(tile[idx] >> shift) & 255u;
      atomicAdd(&hist[d], 1u);
    }
  }
  __syncthreads();
  bh[(size_t)t * numBlocks + blk] = hist[t];
  if (hist[t]) atomicAdd(&dt[t], hist[t]);
}

// Per-digit exclusive scan across blocks (digit-major layout). 256 blocks.
__global__ void k_scan_blockhist(u32* __restrict__ bh, int numBlocks) {
  __shared__ u32 s[TPB];
  __shared__ float sp[TPB];
  int d = blockIdx.x, t = threadIdx.x;
  u32 carry = 0u;
  for (int b0 = 0; b0 < numBlocks; b0 += TPB) {
    int i = b0 + t;
    u32 v = (i < numBlocks) ? bh[(size_t)d * numBlocks + i] : 0u;
    u32 incl = block_incl_scan(v, s, sp);
    if (i < numBlocks) bh[(size_t)d * numBlocks + i] = carry + incl - v;
    u32 tot = s[TPB - 1];
    __syncthreads();
    carry += tot;
  }
}

__global__ void k_digit_base(const u32* __restrict__ totals, u32* __restrict__ base) {
  __shared__ u32 s[TPB];
  __shared__ float sp[TPB];
  int t = threadIdx.x;
  u32 v = totals[t];
  u32 incl = block_incl_scan(v, s, sp);
  base[t] = incl - v;
}

// ------------------------------------------------------ radix: scatter
// Tile fetched via the Tensor Data Mover (1D descriptor); stable rank via
// wave32 ballots (8 waves/block), wave-order prefix in LDS.
__global__ void k_radix_scatter(const u64* __restrict__ in, u64* __restrict__ out,
                                const u32* __restrict__ bh, const u32* __restrict__ digitBase,
                                int E, int shift, int numBlocks) {
  __shared__ u64 tile[KPB];
  __shared__ u32 run[RADIX];
  __shared__ u32 wdc[8 * RADIX];
  __shared__ u32 woff[8 * RADIX];
  int t = threadIdx.x, blk = blockIdx.x;
  int wave = t >> 5, lane = t & 31;
  int base = blk * KPB;

  run[t] = digitBase[t] + bh[(size_t)t * numBlocks + blk];

  if (wave == 0) {
    int cnt = E - base; if (cnt > KPB) cnt = KPB;
    u64 ga  = (u64)(uintptr_t)(in + base);
    u32 lds = (u32)(uintptr_t)(&tile[0]);
    u32 c   = (u32)cnt;
    su4 g0; su8 g1;
    g0[0] = rfl(1u);                                           // count=1 valid
    g0[1] = rfl(lds);                                          // lds_addr
    g0[2] = rfl((u32)ga);                                      // global_addr lo
    g0[3] = rfl(((u32)(ga >> 32) & 0x01FFFFFFu) | (2u << 30)); // addr hi | type=2
    g1[0] = rfl(3u << 16);                 // wg_mask=0, data_size=3 (8B)
    g1[1] = rfl((c & 0xFFFFu) << 16);      // tensor_dim0 lo16
    g1[2] = rfl((c >> 16) | (1u << 16));   // tensor_dim0 hi | tensor_dim1=1 lo
    g1[3] = rfl((c & 0xFFFFu) << 16);      // tensor_dim1 hi=0 | tile_dim0=cnt
    g1[4] = rfl(1u);                       // tile_dim1=1, tile_dim2=0
    g1[5] = rfl(c);                        // tensor_dim0_stride lo32
    g1[6] = rfl(0u);
    g1[7] = rfl(0u);
    asm volatile("tensor_load_to_lds %0, %1" :: "s"(g0), "s"(g1) : "memory");
    __builtin_amdgcn_s_wait_tensorcnt(0);
  }
  __syncthreads();

  #pragma unroll
  for (int j = 0; j < 4; j++) {
    int idx = j * TPB + t;
    bool valid = (base + idx) < E;
    u64 key = valid ? tile[idx] : 0ull;
    u32 d = (u32)(key >> shift) & 255u;

    u32 m = __builtin_amdgcn_ballot_w32(valid);
    #pragma unroll
    for (int b = 0; b < 8; b++) {
      u32 bb = __builtin_amdgcn_ballot_w32(((d >> b) & 1u) != 0u);
      m &= ((d >> b) & 1u) ? bb : ~bb;
    }
    u32 lower = m & ((1u << lane) - 1u);
    u32 rank  = (u32)__popc(lower);
    u32 cnt2  = (u32)__popc(m);
    bool leader = valid && (lower == 0u);

    for (int k2 = t; k2 < 8 * RADIX; k2 += TPB) wdc[k2] = 0u;
    __syncthreads();
    if (leader) wdc[wave * RADIX + d] = cnt2;
    __syncthreads();
    u32 acc = run[t];
    #pragma unroll
    for (int w = 0; w < 8; w++) { woff[w * RADIX + t] = acc; acc += wdc[w * RADIX + t]; }
    run[t] = acc;
    __syncthreads();
    if (valid) out[woff[wave * RADIX + d] + rank] = key;
    __syncthreads();
  }
}

// ----------------------------------------------------------- CSR build
__global__ void k_unpack(const u64* __restrict__ keys, int* __restrict__ srow,
                         int* __restrict__ scol, u32* __restrict__ deg, int E) {
  int i = blockIdx.x * TPB + threadIdx.x;
  if (i < E) {
    u64 k = keys[i];
    u32 r = (u32)(k >> 20);
    u32 c = (u32)(k & 0xFFFFFu);
    srow[i] = (int)r; scol[i] = (int)c;
    atomicAdd(&deg[r], 1u);
  }
}

// -------------------------------------------------- generic exclusive scan
__global__ void k_scan_blocks(const u32* __restrict__ in, u32* __restrict__ out,
                              u32* __restrict__ partials, int n) {
  __shared__ u32 s[TPB];
  __shared__ float sp[TPB];
  int t = threadIdx.x;
  int base = blockIdx.x * KPB + t * 4;
  u32 v0 = (base + 0 < n) ? in[base + 0] : 0u;
  u32 v1 = (base + 1 < n) ? in[base + 1] : 0u;
  u32 v2 = (base + 2 < n) ? in[base + 2] : 0u;
  u32 v3 = (base + 3 < n) ? in[base + 3] : 0u;
  u32 tsum = v0 + v1 + v2 + v3;
  u32 incl = block_incl_scan(tsum, s, sp);
  u32 r = incl - tsum;
  if (base + 0 < n) out[base + 0] = r; r += v0;
  if (base + 1 < n) out[base + 1] = r; r += v1;
  if (base + 2 < n) out[base + 2] = r; r += v2;
  if (base + 3 < n) out[base + 3] = r;
  if (t == TPB - 1) partials[blockIdx.x] = incl;
}

__global__ void k_scan_serial(u32* __restrict__ a, int n, u32* __restrict__ total) {
  __shared__ u32 s[TPB];
  __shared__ float sp[TPB];
  int t = threadIdx.x;
  u32 carry = 0u;
  for (int b0 = 0; b0 < n; b0 += TPB) {
    int i = b0 + t;
    u32 v = (i < n) ? a[i] : 0u;
    u32 incl = block_incl_scan(v, s, sp);
    if (i < n) a[i] = carry + incl - v;
    u32 tot = s[TPB - 1];
    __syncthreads();
    carry += tot;
  }
  if (t == 0) *total = carry;
}

__global__ void k_scan_add(u32* __restrict__ a, const u32* __restrict__ partials, int n) {
  int t = threadIdx.x;
  int base = blockIdx.x * KPB + t * 4;
  u32 p = partials[blockIdx.x];
  #pragma unroll
  for (int j = 0; j < 4; j++)
    if (base + j < n) a[base + j] += p;
}

// ----------------------------------------------------------- random walk
__global__ void k_walk(const u32* __restrict__ deg, const u32* __restrict__ rowptr,
                       const int* __restrict__ scol, u8* __restrict__ hits,
                       int N, int S) {
  int i = blockIdx.x * TPB + threadIdx.x;
  if (i >= S) return;
  // pseudo-random permutation start node: 579131 coprime to 1e6
  u32 cur = (u32)((579131ull * (u64)i + 123457ull) % (u64)N);
  bool active = true;
  #pragma unroll
  for (int s = 0; s < 3; s++) {
    u32 d = deg[cur];
    bool act = active && (d > 0u);
    if (act) {
      u32 h = hmix((u32)i ^ (0x9E3779B9u * (u32)(s + 1)));
      float u = (float)(h >> 8) * (1.0f / 16777216.0f);
      u32 off = (u32)(u * (float)d);
      if (off >= d) off = d - 1u;
      u32 nxt = (u32)scol[rowptr[cur] + off];
      hits[cur] = 1u;          // eid = current node (bug-faithful), valid only
      cur = nxt;
    }
    active = act;
  }
}

// --------------------------------------------------------- mask + compact
__global__ void k_flags(const u8* __restrict__ hits, u32* __restrict__ keep, int E) {
  int i = blockIdx.x * TPB + threadIdx.x;
  if (i < E) keep[i] = hits[i] ? 0u : 1u;
}

__global__ void k_compact(const int* __restrict__ srow, const int* __restrict__ scol,
                          const u8* __restrict__ hits, const u32* __restrict__ pscan,
                          float* __restrict__ out, int E) {
  int i = blockIdx.x * TPB + threadIdx.x;
  if (i >= E) return;
  float* remR = out;
  float* remC = out + (size_t)E;
  float* mR   = out + 2 * (size_t)E;
  float* mC   = out + 3 * (size_t)E;
  float* msk  = out + 4 * (size_t)E;
  bool keep = (hits[i] == 0u);
  u32 pk = pscan[i];
  msk[i] = keep ? 1.0f : 0.0f;
  if (keep) { remR[pk] = (float)srow[i]; remC[pk] = (float)scol[i]; }
  else      { u32 pm = (u32)i - pk; mR[pm] = (float)srow[i]; mC[pm] = (float)scol[i]; }
}

__global__ void k_fill(float* __restrict__ out, const u32* __restrict__ nKeep, int E) {
  int i = blockIdx.x * TPB + threadIdx.x;
  if (i >= E) return;
  u32 nk = *nKeep;
  float* remR = out;
  float* remC = out + (size_t)E;
  float* mR   = out + 2 * (size_t)E;
  float* mC   = out + 3 * (size_t)E;
  if ((u32)i >= nk)            { remR[i] = -1.0f; remC[i] = -1.0f; }
  if ((u32)i >= (u32)E - nk)   { mR[i]   = -1.0f; mC[i]   = -1.0f; }
}

// ----------------------------------------------------------------- driver
extern "C" void kernel_launch(void* const* d_in, const int* in_sizes, int n_in,
                              void* d_out, int out_size, void* d_ws, size_t ws_size,
                              hipStream_t stream) {
  const int E = in_sizes[0] / 2;
  const int N = 1000000;                 // reference setup_inputs()
  const int S = 700000;                  // round(N * 0.7) * WALKS_PER_NODE

  const int* row = (const int*)d_in[0];
  const int* col = row + E;
  float* out = (float*)d_out;

  const int nbSort = (E + KPB - 1) / KPB;    // 15625
  const int nbN    = (N + KPB - 1) / KPB;    // 977
  const int gridE  = (E + TPB - 1) / TPB;

  // workspace layout (~300MB)
  char* w = (char*)d_ws;
  u64* keysA       = (u64*)w;  w += (size_t)E * 8;
  u64* keysB       = (u64*)w;  w += (size_t)E * 8;
  u32* bh          = (u32*)w;  w += (size_t)RADIX * nbSort * 4;
  u32* digitTotals = (u32*)w;  w += RADIX * 4;
  u32* digitBase   = (u32*)w;  w += RADIX * 4;
  u32* deg         = (u32*)w;  w += (size_t)N * 4;
  u32* rowptr      = (u32*)w;  w += (size_t)N * 4;
  u32* partials    = (u32*)w;  w += 16384 * 4;
  u32* nKeep       = (u32*)w;  w += 256;
  u8*  hits        = (u8*)w;   w += (size_t)E;
  // overlays (valid after the sort finishes):
  int* srow = (int*)keysA;
  int* scol = ((int*)keysA) + E;
  u32* keepScan = (u32*)keysB;

  dim3 blk(TPB);

  // 1) pack 40-bit keys
  k_pack<<<gridE, blk, 0, stream>>>(row, col, keysA, E);

  // 2) LSD radix sort, 5 x 8-bit passes
  u64* src = keysA; u64* dst = keysB;
  for (int p = 0; p < 5; p++) {
    int shift = p * 8;
    k_zero<<<1, blk, 0, stream>>>(digitTotals, RADIX);
    k_radix_hist<<<nbSort, blk, 0, stream>>>(src, bh, digitTotals, E, shift, nbSort);
    k_scan_blockhist<<<RADIX, blk, 0, stream>>>(bh, nbSort);
    k_digit_base<<<1, blk, 0, stream>>>(digitTotals, digitBase);
    k_radix_scatter<<<nbSort, blk, 0, stream>>>(src, dst, bh, digitBase, E, shift, nbSort);
    u64* tmp = src; src = dst; dst = tmp;
  }
  // src == keysB holds sorted keys; keysA region is free -> srow/scol

  // 3) CSR
  k_zero<<<(N + TPB - 1) / TPB, blk, 0, stream>>>(deg, N);
  k_unpack<<<gridE, blk, 0, stream>>>(src, srow, scol, deg, E);
  k_scan_blocks<<<nbN, blk, 0, stream>>>(deg, rowptr, partials, N);
  k_scan_serial<<<1, blk, 0, stream>>>(partials, nbN, nKeep);   // nKeep reused as scratch
  k_scan_add<<<nbN, blk, 0, stream>>>(rowptr, partials, N);

  // 4) walks + hit scatter
  k_zero<<<(E / 4 + TPB - 1) / TPB, blk, 0, stream>>>((u32*)hits, E / 4);
  k_walk<<<(S + TPB - 1) / TPB, blk, 0, stream>>>(deg, rowptr, scol, hits, N, S);

  // 5) keep flags + 16M exclusive scan (keysB region reused)
  k_flags<<<gridE, blk, 0, stream>>>(hits, keepScan, E);
  k_scan_blocks<<<nbSort, blk, 0, stream>>>(keepScan, keepScan, partials, E);
  k_scan_serial<<<1, blk, 0, stream>>>(partials, nbSort, nKeep);
  k_scan_add<<<nbSort, blk, 0, stream>>>(keepScan, partials, E);

  // 6) stable compaction + edge_mask + -1 tails
  k_compact<<<gridE, blk, 0, stream>>>(srow, scol, hits, keepScan, out, E);
  k_fill<<<gridE, blk, 0, stream>>>(out, nKeep, E);

  (void)n_in; (void)out_size; (void)ws_size;
}